// FocalLoss_65283502899339
// MI455X (gfx1250) — compile-verified
//
#include <hip/hip_runtime.h>
#include <math.h>

#define ALPHA     0.25f
#define EPS_F     1e-6f
#define ONE_M_EPS (1.0f - 1e-6f)
#define DL        (1.0f / 9.0f)     // smooth-L1 divide line
#define HALF_DL   (0.5f / 9.0f)
#define INV2DL    4.5f              // 0.5 / DL

typedef float v2f __attribute__((ext_vector_type(2)));
typedef float v8f __attribute__((ext_vector_type(8)));

// ---------------------------------------------------------------------------
// Wave32 reduction via V_WMMA_F32_16X16X4_F32 with B = ones:
//   A-matrix (16x4 f32, 2 VGPRs/lane): lanes 0-15 hold K=0 (v0) / K=1 (v1),
//   lanes 16-31 hold K=2 / K=3.  Put each lane's partial in v0, zero in v1.
//   D[m][n] = sum_k A[m][k] = s_m + s_{m+16}, identical for every column n.
//   Each lane then sums its 8 C/D VGPRs (rows M= lane<16 ? 0..7 : 8..15) and
//   one shfl_xor(16) combines the two halves -> full wave sum in all lanes.
// EXEC must be all ones: call only from fully converged code.
// ---------------------------------------------------------------------------
__device__ __forceinline__ float wave_reduce_wmma(float s) {
  v2f a; a[0] = s;    a[1] = 0.0f;
  v2f b; b[0] = 1.0f; b[1] = 1.0f;
  v8f c = {};
  c = __builtin_amdgcn_wmma_f32_16x16x4_f32(
      /*neg_a=*/false, a, /*neg_b=*/false, b,
      /*c_mod=*/(short)0, c, /*reuse_a=*/false, /*reuse_b=*/false);
  float t = (c[0] + c[1]) + (c[2] + c[3]) + (c[4] + c[5]) + (c[6] + c[7]);
  t += __shfl_xor(t, 16, 32);
  return t;
}

__device__ __forceinline__ float bg_term(float p) {
  p = fminf(fmaxf(p, EPS_F), ONE_M_EPS);
  return 0.75f * (-logf(1.0f - p)) * p * p;
}

// ---------------------------------------------------------------------------
// Kernel 1: per-anchor matching + focal correction + smooth-L1 regression.
// grid = (ceil(A/256), N), block = 256. GT boxes/labels staged in LDS.
// ws layout (floats): [0,N) bg, [N,2N) corr, [2N,3N) regsum ; ints at [3N,4N)
// ---------------------------------------------------------------------------
__global__ void __launch_bounds__(256)
match_correct_kernel(const float* __restrict__ cls,
                     const float* __restrict__ regs,
                     const float* __restrict__ anchors,
                     const float* __restrict__ gt_boxes,
                     const int*   __restrict__ gt_labels,
                     float* __restrict__ wsf, int* __restrict__ wsi,
                     int A, int G, int C) {
  __shared__ float gb[64 * 4];
  __shared__ int   gl[64];
  __shared__ float s_corr, s_reg;
  __shared__ int   s_p;

  const int img = blockIdx.y;
  for (int j = threadIdx.x; j < G * 4; j += blockDim.x)
    gb[j] = gt_boxes[(size_t)img * G * 4 + j];
  for (int j = threadIdx.x; j < G; j += blockDim.x)
    gl[j] = gt_labels[(size_t)img * G + j];
  if (threadIdx.x == 0) { s_corr = 0.0f; s_reg = 0.0f; s_p = 0; }
  __syncthreads();

  float corr = 0.0f, regsum = 0.0f;
  int   pcnt = 0;

  const int a = blockIdx.x * blockDim.x + threadIdx.x;
  if (a < A) {
    const float ax1 = anchors[a * 4 + 0];
    const float ay1 = anchors[a * 4 + 1];
    const float ax2 = anchors[a * 4 + 2];
    const float ay2 = anchors[a * 4 + 3];
    const float areaA = (ax2 - ax1) * (ay2 - ay1);

    float best = -1.0f;
    int   m    = 0;
    for (int g = 0; g < G; ++g) {
      const float bx1 = gb[g * 4 + 0], by1 = gb[g * 4 + 1];
      const float bx2 = gb[g * 4 + 2], by2 = gb[g * 4 + 3];
      const float w  = fmaxf(fminf(ax2, bx2) - fmaxf(ax1, bx1), 0.0f);
      const float h  = fmaxf(fminf(ay2, by2) - fmaxf(ay1, by1), 0.0f);
      const float inter = w * h;
      const float areaB = (bx2 - bx1) * (by2 - by1);
      const float iou = inter / (areaA + areaB - inter);
      if (iou > best) { best = iou; m = g; }   // strict > == argmax first-max
    }

    const bool pos = best >= 0.5f;
    const bool ign = (best >= 0.4f) && !pos;
    if (pos || ign) {
      const int lbl = gl[m];
      float p = cls[((size_t)img * A + a) * C + lbl];
      p = fminf(fmaxf(p, EPS_F), ONE_M_EPS);
      const float omp = 1.0f - p;
      const float bg  = 0.75f * (-logf(omp)) * p * p;
      if (ign) {
        corr -= bg;                       // drop the masked (-1) element
      } else {
        corr += ALPHA * (-logf(p)) * omp * omp - bg;   // swap bg -> fg term
        pcnt  = 1;
        // regression targets (encode matched gt vs anchor)
        const float bx1 = gb[m * 4 + 0], by1 = gb[m * 4 + 1];
        const float bx2 = gb[m * 4 + 2], by2 = gb[m * 4 + 3];
        const float wa = ax2 - ax1, ha = ay2 - ay1;
        const float cxa = ax1 + 0.5f * wa, cya = ay1 + 0.5f * ha;
        const float wg = bx2 - bx1, hg = by2 - by1;
        const float cxg = bx1 + 0.5f * wg, cyg = by1 + 0.5f * hg;
        float t[4];
        t[0] = (cxg - cxa) / wa;
        t[1] = (cyg - cya) / ha;
        t[2] = logf(wg / wa);
        t[3] = logf(hg / ha);
        const float* r = regs + ((size_t)img * A + a) * 4;
        #pragma unroll
        for (int k = 0; k < 4; ++k) {
          const float diff = fabsf(r[k] - t[k]);
          regsum += (diff < DL) ? INV2DL * diff * diff : diff - HALF_DL;
        }
      }
    }
  }

  if (corr   != 0.0f) atomicAdd(&s_corr, corr);
  if (regsum != 0.0f) atomicAdd(&s_reg, regsum);
  if (pcnt)           atomicAdd(&s_p, 1);
  __syncthreads();
  if (threadIdx.x == 0) {
    if (s_corr != 0.0f) atomicAdd(&wsf[1 * gridDim.y + img], s_corr);
    if (s_reg  != 0.0f) atomicAdd(&wsf[2 * gridDim.y + img], s_reg);
    if (s_p)            atomicAdd(&wsi[img], s_p);
  }
}

// ---------------------------------------------------------------------------
// Kernel 2: streaming background focal sum over [A*C] per image (float4).
// grid = (BX, N), block = 256. This is the bandwidth-bound pass (~141 MB).
// ---------------------------------------------------------------------------
__global__ void __launch_bounds__(256)
bg_stream_kernel(const float* __restrict__ cls, float* __restrict__ ws_bg,
                 int AC4) {
  const int img = blockIdx.y;
  const float4* __restrict__ base =
      reinterpret_cast<const float4*>(cls) + (size_t)img * AC4;

  const int stride = gridDim.x * blockDim.x;
  float acc = 0.0f;
  for (int e = blockIdx.x * blockDim.x + threadIdx.x; e < AC4; e += stride) {
    __builtin_prefetch(base + e + stride, 0, 0);   // global_prefetch_b8
    const float4 v = base[e];                      // global_load_b128
    acc += bg_term(v.x) + bg_term(v.y) + bg_term(v.z) + bg_term(v.w);
  }

  // fully converged here: wave32 WMMA reduction is legal (EXEC all ones)
  const float wsum = wave_reduce_wmma(acc);

  __shared__ float warp_s[8];
  const int lane = threadIdx.x & 31;
  const int wid  = threadIdx.x >> 5;
  if (lane == 0) warp_s[wid] = wsum;
  __syncthreads();
  if (threadIdx.x == 0) {
    float t = 0.0f;
    const int nw = blockDim.x >> 5;
    for (int w = 0; w < nw; ++w) t += warp_s[w];
    atomicAdd(&ws_bg[img], t);
  }
}

// ---------------------------------------------------------------------------
// Kernel 3: finalize two scalars.
// ---------------------------------------------------------------------------
__global__ void finalize_kernel(const float* __restrict__ wsf,
                                const int* __restrict__ wsi,
                                float* __restrict__ out, int N) {
  if (blockIdx.x == 0 && threadIdx.x == 0) {
    float cl = 0.0f, rl = 0.0f;
    for (int i = 0; i < N; ++i) {
      const int   P  = wsi[i];
      const float fs = wsf[i] + wsf[N + i];           // bg + correction
      cl += fs / (float)max(P, 1);
      rl += (P > 0) ? wsf[2 * N + i] / (float)max(4 * P, 1) : 0.0f;
    }
    out[0] = cl / (float)N;
    out[1] = rl / (float)N;
  }
}

// ---------------------------------------------------------------------------
extern "C" void kernel_launch(void* const* d_in, const int* in_sizes, int n_in,
                              void* d_out, int out_size, void* d_ws, size_t ws_size,
                              hipStream_t stream) {
  const float* cls     = (const float*)d_in[0];   // [N, A, C]
  const float* regs    = (const float*)d_in[1];   // [N, A, 4]
  const float* anchors = (const float*)d_in[2];   // [A, 4]
  const float* gtb     = (const float*)d_in[3];   // [N, G, 4]
  const int*   gtl     = (const int*)d_in[4];     // [N, G]
  float*       out     = (float*)d_out;           // [class_loss, reg_loss]

  const int A = in_sizes[2] / 4;
  const int N = in_sizes[1] / (A * 4);
  const int G = in_sizes[4] / N;
  const int C = in_sizes[0] / (N * A);

  float* wsf = (float*)d_ws;            // [0,N) bg | [N,2N) corr | [2N,3N) reg
  int*   wsi = (int*)d_ws + 3 * N;      // [3N,4N) positive counts

  hipMemsetAsync(d_ws, 0, (size_t)(4 * N) * sizeof(float), stream);

  dim3 b1(256), g1((A + 255) / 256, N);
  match_correct_kernel<<<g1, b1, 0, stream>>>(cls, regs, anchors, gtb, gtl,
                                              wsf, wsi, A, G, C);

  const int AC4 = (A * C) / 4;          // A*C divisible by 4 for this problem
  int bx = (AC4 + 255) / 256;
  if (bx > 1024) bx = 1024;
  dim3 b2(256), g2(bx, N);
  bg_stream_kernel<<<g2, b2, 0, stream>>>(cls, wsf, AC4);

  finalize_kernel<<<1, 32, 0, stream>>>(wsf, wsi, out, N);
}